// AttnModel_1580547967294
// MI455X (gfx1250) — compile-verified
//
#include <hip/hip_runtime.h>
#include <math.h>

typedef __attribute__((ext_vector_type(2))) float v2f;
typedef __attribute__((ext_vector_type(8))) float v8f;

#if __has_builtin(__builtin_amdgcn_tanhf)
__device__ __forceinline__ float fast_tanh(float x) { return __builtin_amdgcn_tanhf(x); }
#else
__device__ __forceinline__ float fast_tanh(float x) { return tanhf(x); }
#endif

// ---------------------------------------------------------------------------
// Generic fp32 WMMA GEMM:  C[z] = act( A[z](MxK) @ B[z](KxN) + bias )
// One wave computes one 16x16 C tile using V_WMMA_F32_16X16X4_F32, K step 4.
// VGPR layouts per CDNA5 ISA 7.12.2:
//   A 16x4 : lane holds A[m = lane&15][k0 + 2*(lane>>4) + {0,1}] in {a.x,a.y}
//   B 4x16 : lane holds B[k0 + 2*(lane>>4) + {0,1}][n = lane&15] in {b.x,b.y}
//   C 16x16: VGPR j -> row (j + 8*(lane>>4)), col (lane&15)
// ---------------------------------------------------------------------------
__global__ __launch_bounds__(128) void wmma_gemm_f32(
    const float* __restrict__ A, const float* __restrict__ Bm,
    const float* __restrict__ bias, float* __restrict__ C,
    int M, int N, int K,
    long sA, long sB, long sC, int act)
{
  const int lane = threadIdx.x;          // 0..31, wave32
  const int m0 = blockIdx.x * 16;
  const int n0 = (blockIdx.y * 4 + threadIdx.y) * 16;
  const float* Ab = A  + (long)blockIdx.z * sA;
  const float* Bb = Bm + (long)blockIdx.z * sB;
  float*       Cb = C  + (long)blockIdx.z * sC;

  const int mr = lane & 15;
  const int hi = lane >> 4;              // which half-wave
  const int kh = hi * 2;                 // K sub-offset 0 or 2

  v8f c = {};
  for (int k0 = 0; k0 < K; k0 += 4) {
    v2f a, b;
    const float* ap = Ab + (long)(m0 + mr) * K + (k0 + kh);
    a.x = ap[0];
    a.y = ap[1];
    const float* bp = Bb + (long)(k0 + kh) * N + (n0 + mr);
    b.x = bp[0];
    b.y = bp[N];
    // (neg_a, A, neg_b, B, c_mod, C, reuse_a, reuse_b)
    c = __builtin_amdgcn_wmma_f32_16x16x4_f32(false, a, false, b, (short)0, c,
                                              false, false);
  }

#pragma unroll
  for (int j = 0; j < 8; ++j) {
    const int row = m0 + j + 8 * hi;
    const int col = n0 + mr;
    float v = c[j];
    if (bias) v += bias[col];
    if (act)  v = (v > 0.0f) ? v : 0.01f * v;   // leaky_relu(0.01)
    Cb[(long)row * N + col] = v;
  }
}

// ---------------------------------------------------------------------------
// Additive-attention score + masked softmax, G=4 output rows per block.
// One block per (b, o-tile of 4) -> 128 blocks, 256 threads = 8 waves.
// score[g][i] = sum_h w2[h]*tanh(hq[o0+g,h] + hkb[i,h]) + b2, softmax over i.
// hkb already contains hk + b1. Each hk row load (2x global_load_b128/lane)
// is reused for 4 tanh rows; hq/w2 slices live in per-lane registers.
// Lane h-slice: h in {4*lane..4*lane+3} U {128+4*lane..128+4*lane+3}.
// ---------------------------------------------------------------------------
#define OG 4
__global__ __launch_bounds__(256) void score_softmax(
    const float* __restrict__ hq, const float* __restrict__ hkb,
    const float* __restrict__ w2, const float* __restrict__ b2,
    const int* __restrict__ mask, float* __restrict__ attn)
{
  __shared__ float sc_s[OG][512];
  __shared__ float red_s[256];

  const int b  = blockIdx.x >> 4;          // OUT/OG = 16 o-tiles per batch
  const int o0 = (blockIdx.x & 15) * OG;
  const int t  = threadIdx.x;
  const int wave = t >> 5;
  const int lane = t & 31;

  // register-resident operand slices
  float w2r[8], hqr[OG][8];
  *(float4*)&w2r[0] = *(const float4*)(w2 + 4 * lane);
  *(float4*)&w2r[4] = *(const float4*)(w2 + 128 + 4 * lane);
#pragma unroll
  for (int g = 0; g < OG; ++g) {
    const float* hp = hq + (long)(b * 64 + o0 + g) * 256;
    *(float4*)&hqr[g][0] = *(const float4*)(hp + 4 * lane);
    *(float4*)&hqr[g][4] = *(const float4*)(hp + 128 + 4 * lane);
  }

  const float bias2 = b2[0];
  const float* hkB = hkb + (long)b * 512 * 256;

  for (int i = wave; i < 512; i += 8) {
    const float* row = hkB + (long)i * 256 + 4 * lane;
    float rr[8];
    *(float4*)&rr[0] = *(const float4*)(row);
    *(float4*)&rr[4] = *(const float4*)(row + 128);

    float s[OG];
#pragma unroll
    for (int g = 0; g < OG; ++g) {
      float acc = 0.0f;
#pragma unroll
      for (int j = 0; j < 8; ++j)
        acc = fmaf(w2r[j], fast_tanh(hqr[g][j] + rr[j]), acc);
      s[g] = acc;
    }
    // butterfly reduce across the wave (wave32)
#pragma unroll
    for (int m = 16; m > 0; m >>= 1) {
#pragma unroll
      for (int g = 0; g < OG; ++g) s[g] += __shfl_xor(s[g], m, 32);
    }
    if (lane == 0) {
      const bool msk = (mask[b * 512 + i] != 0);
#pragma unroll
      for (int g = 0; g < OG; ++g)
        sc_s[g][i] = msk ? -INFINITY : (s[g] + bias2);
    }
  }
  __syncthreads();

  // masked softmax per output row
  for (int g = 0; g < OG; ++g) {
    float mx = fmaxf(sc_s[g][t], sc_s[g][t + 256]);
    red_s[t] = mx;
    __syncthreads();
    for (int sft = 128; sft > 0; sft >>= 1) {
      if (t < sft) red_s[t] = fmaxf(red_s[t], red_s[t + sft]);
      __syncthreads();
    }
    mx = red_s[0];
    __syncthreads();

    const float p0 = expf(sc_s[g][t] - mx);
    const float p1 = expf(sc_s[g][t + 256] - mx);
    red_s[t] = p0 + p1;
    __syncthreads();
    for (int sft = 128; sft > 0; sft >>= 1) {
      if (t < sft) red_s[t] += red_s[t + sft];
      __syncthreads();
    }
    const float inv = 1.0f / red_s[0];
    __syncthreads();   // red_s reused next g

    const long base = (long)(b * 64 + o0 + g) * 512;
    attn[base + t]       = p0 * inv;
    attn[base + t + 256] = p1 * inv;
  }
}

// ---------------------------------------------------------------------------
// Launch: 5 kernels, all on `stream`, graph-capture safe.
// ---------------------------------------------------------------------------
extern "C" void kernel_launch(void* const* d_in, const int* in_sizes, int n_in,
                              void* d_out, int out_size, void* d_ws, size_t ws_size,
                              hipStream_t stream) {
  const float* q    = (const float*)d_in[0];   // (8,64,256)
  const float* k    = (const float*)d_in[1];   // (8,512,256)  (also v)
  const int*   mask = (const int*)  d_in[2];   // (8,512)
  const float* w1   = (const float*)d_in[3];   // (512,256): wq rows 0..255, wk rows 256..511
  const float* b1   = (const float*)d_in[4];   // (256)
  const float* w2   = (const float*)d_in[5];   // (256,1)
  const float* b2   = (const float*)d_in[6];   // (1)
  const float* fw_w = (const float*)d_in[7];   // (256,256)
  const float* fw_b = (const float*)d_in[8];   // (256)

  float* out  = (float*)d_out;                 // (8,64,256) = 131072
  float* attn = (float*)d_out + 131072;        // (8,64,512) = 262144

  float* hq   = (float*)d_ws;                  // 512*256
  float* hkb  = hq  + 512 * 256;               // 4096*256  (hk + b1)
  float* out0 = hkb + 4096 * 256;              // 512*256

  const dim3 blk(32, 4, 1);                    // 4 waves, one 16x16 tile each

  // hq = q @ wq                 (M=512, N=256, K=256)
  wmma_gemm_f32<<<dim3(512 / 16, 256 / 64, 1), blk, 0, stream>>>(
      q, w1, nullptr, hq, 512, 256, 256, 0, 0, 0, 0);

  // hkb = k @ wk + b1           (M=4096, N=256, K=256)
  wmma_gemm_f32<<<dim3(4096 / 16, 256 / 64, 1), blk, 0, stream>>>(
      k, w1 + 256 * 256, b1, hkb, 4096, 256, 256, 0, 0, 0, 0);

  // score + masked softmax -> attn (second output), 128 blocks x 256 thr
  score_softmax<<<dim3(8 * 16), dim3(256), 0, stream>>>(
      hq, hkb, w2, b2, mask, attn);

  // out0 = attn @ v, batched over b   (M=64, N=256, K=512, 8 batches)
  wmma_gemm_f32<<<dim3(64 / 16, 256 / 64, 8), blk, 0, stream>>>(
      attn, k, nullptr, out0, 64, 256, 512,
      (long)64 * 512, (long)512 * 256, (long)64 * 256, 0);

  // out = leaky_relu(out0 @ fw_w + fw_b)   (M=512, N=256, K=256)
  wmma_gemm_f32<<<dim3(512 / 16, 256 / 64, 1), blk, 0, stream>>>(
      out0, fw_w, fw_b, out, 512, 256, 256, 0, 0, 0, 1);
}